// ConvCRF_40561671143560
// MI455X (gfx1250) — compile-verified
//
#include <hip/hip_runtime.h>

typedef __attribute__((ext_vector_type(2))) float v2f;
typedef __attribute__((ext_vector_type(8))) float v8f;
typedef __attribute__((ext_vector_type(4))) unsigned u32x4;
typedef __attribute__((ext_vector_type(8))) int i32x8;
typedef __attribute__((ext_vector_type(4))) int i32x4;

constexpr int kH = 160, kW = 160, kNPIX = kH * kW;
constexpr int kC = 19;          // classes
constexpr int kB = 2;           // batch
constexpr int kSPAN = 3, kFS = 7, kTAPS = 49;
constexpr int kNITER = 5;
constexpr float kINV_ALPHA = 1.0f / 160.0f;   // theta_alpha
constexpr float kINV_BG2   = 1.0f / 9.0f;     // 1/theta_beta^2 == 1/theta_gamma^2 (both 3.0)
constexpr float kEPS = 1e-20f;

// ---- workspace layout (float offsets) ----
constexpr size_t OFF_A   = 0;                        // [2][32][20] combined weights (CM@SW, CM@BW), zero-padded
constexpr size_t OFF_INS = OFF_A   + 2 * 32 * 20;    // [NPIX]       1/(spatial norm + eps)
constexpr size_t OFF_INB = OFF_INS + kNPIX;          // [B][NPIX]    1/(bilateral norm + eps)
constexpr size_t OFF_FB  = OFF_INB + (size_t)kB * kNPIX;            // [B][49][NPIX] bilateral filter
constexpr size_t OFF_U   = OFF_FB  + (size_t)kB * kTAPS * kNPIX;    // [B][C][NPIX] unaries NCHW
constexpr size_t OFF_Q   = OFF_U   + (size_t)kB * kC * kNPIX;       // [B][C][NPIX]
constexpr size_t OFF_YS  = OFF_Q   + (size_t)kB * kC * kNPIX;       // [B][C][NPIX] (CM@SW)@softmax
constexpr size_t OFF_YB  = OFF_YS  + (size_t)kB * kC * kNPIX;       // [B][C][NPIX] (CM@BW)@softmax

// ---------------------------------------------------------------------------
// K0: combined 19x19 weight products, zero-padded into [2][32][20]
// ---------------------------------------------------------------------------
__global__ void prep_weights(const float* __restrict__ sw, const float* __restrict__ bw,
                             const float* __restrict__ cm, float* __restrict__ A) {
    for (int i = threadIdx.x; i < 2 * 32 * 20; i += blockDim.x) {
        int mat = i / 640, rem = i % 640, m = rem / 20, k = rem % 20;
        float v = 0.f;
        if (m < kC && k < kC) {
            const float* wsrc = mat ? bw : sw;
            for (int j = 0; j < kC; ++j) v += cm[m * kC + j] * wsrc[j * kC + k];
        }
        A[i] = v;
    }
}

// ---------------------------------------------------------------------------
// K1: NHWC unaries -> NCHW u and q(=u)
// ---------------------------------------------------------------------------
__global__ void init_q(const float* __restrict__ un, float* __restrict__ u, float* __restrict__ q) {
    int idx = blockIdx.x * blockDim.x + threadIdx.x;
    if (idx >= kB * kC * kNPIX) return;
    int p = idx % kNPIX, c = (idx / kNPIX) % kC, b = idx / (kNPIX * kC);
    float v = un[((size_t)b * kNPIX + p) * kC + c];
    u[idx] = v;
    q[idx] = v;
}

// ---------------------------------------------------------------------------
// K2: bilateral filter + inverse norms (spatial norm from constant tap weights)
// ---------------------------------------------------------------------------
__global__ void build_filters(const float* __restrict__ rgb, float* __restrict__ fb,
                              float* __restrict__ inv_ns, float* __restrict__ inv_nb) {
    int p = blockIdx.x * blockDim.x + threadIdx.x;
    int b = blockIdx.y;
    if (p >= kNPIX) return;
    int h = p / kW, w = p % kW;
    const float* rp = rgb + ((size_t)b * kNPIX + p) * 3;
    float r0 = rp[0] * kINV_ALPHA, g0 = rp[1] * kINV_ALPHA, bl0 = rp[2] * kINV_ALPHA;
    float nb = 0.f, ns = 0.f;
    for (int dx = -kSPAN; dx <= kSPAN; ++dx) {
        for (int dy = -kSPAN; dy <= kSPAN; ++dy) {
            int t = (dx + kSPAN) * kFS + (dy + kSPAN);
            int hh = h + dx, ww = w + dy;
            float sp = (float)(dx * dx + dy * dy) * kINV_BG2;
            float aff = 0.f;
            if (hh >= 0 && hh < kH && ww >= 0 && ww < kW) {
                const float* rq = rgb + ((size_t)b * kNPIX + hh * kW + ww) * 3;
                float dr = r0 - rq[0] * kINV_ALPHA;
                float dg = g0 - rq[1] * kINV_ALPHA;
                float db = bl0 - rq[2] * kINV_ALPHA;
                aff = __expf(-0.5f * (dr * dr + dg * dg + db * db + sp));
                nb += aff;
                ns += __expf(-0.5f * sp);
            }
            fb[((size_t)b * kTAPS + t) * kNPIX + p] = aff;
        }
    }
    inv_nb[(size_t)b * kNPIX + p] = 1.f / (nb + kEPS);
    if (b == 0) inv_ns[p] = 1.f / (ns + kEPS);
}

// ---------------------------------------------------------------------------
// K3: fused batch-axis softmax + GEMM via V_WMMA_F32_16X16X4_F32.
//     Y[2 mats][2 batches](38x51200) = A(38x20-padded) x softmax(q)(20x51200)
//     Each wave: one 16-pixel N tile, 2 M tiles, 5 K chunks, both matrices,
//     both batches -> 40 WMMA per wave. EXEC all-ones (no divergence).
// ---------------------------------------------------------------------------
__global__ __launch_bounds__(256) void gemm_softmax(
    const float* __restrict__ q, const float* __restrict__ A,
    float* __restrict__ ys, float* __restrict__ yb) {
    const int tid  = threadIdx.x;
    const int lane = tid & 31;
    const int wave = tid >> 5;
    const int half = lane >> 4;
    const int lid  = lane & 15;
    const int np   = (blockIdx.x * 8 + wave) * 16;   // pixel base of this N tile

    v8f as_acc[2][2] = {{{}, {}}, {{}, {}}};   // [batch][mtile]
    v8f ab_acc[2][2] = {{{}, {}}, {{}, {}}};

#pragma unroll
    for (int kc = 0; kc < 5; ++kc) {
        // A 16x4 f32 layout: lanes 0-15 hold K=(4kc,4kc+1), lanes 16-31 K=(4kc+2,4kc+3)
        int kr0 = kc * 4 + half * 2;
        int kr1 = kr0 + 1;
        int kr1c = (kr1 < kC) ? kr1 : (kC - 1);   // clamp: keeps loads uniform, EXEC untouched
        // softmax over the batch axis (2 elements), max-subtracted like jax.nn.softmax
        float q0r0 = q[(0 * kC + kr0) * kNPIX + np + lid];
        float q1r0 = q[(1 * kC + kr0) * kNPIX + np + lid];
        float m0 = fmaxf(q0r0, q1r0);
        float e00 = __expf(q0r0 - m0), e10 = __expf(q1r0 - m0);
        float inv0 = __builtin_amdgcn_rcpf(e00 + e10);
        float q0r1 = q[(0 * kC + kr1c) * kNPIX + np + lid];
        float q1r1 = q[(1 * kC + kr1c) * kNPIX + np + lid];
        float m1 = fmaxf(q0r1, q1r1);
        float e01 = __expf(q0r1 - m1), e11 = __expf(q1r1 - m1);
        float inv1 = (kr1 < kC) ? __builtin_amdgcn_rcpf(e01 + e11) : 0.f;  // zero padded K row
        v2f B0, B1;
        B0.x = e00 * inv0; B0.y = e01 * inv1;   // batch 0
        B1.x = e10 * inv0; B1.y = e11 * inv1;   // batch 1
#pragma unroll
        for (int mt = 0; mt < 2; ++mt) {
            int m = mt * 16 + lid;
            v2f As, Ab;
            As.x = A[m * 20 + kr0];        As.y = A[m * 20 + kr1];
            Ab.x = A[640 + m * 20 + kr0];  Ab.y = A[640 + m * 20 + kr1];
            as_acc[0][mt] = __builtin_amdgcn_wmma_f32_16x16x4_f32(
                false, As, false, B0, (short)0, as_acc[0][mt], false, false);
            as_acc[1][mt] = __builtin_amdgcn_wmma_f32_16x16x4_f32(
                false, As, false, B1, (short)0, as_acc[1][mt], false, false);
            ab_acc[0][mt] = __builtin_amdgcn_wmma_f32_16x16x4_f32(
                false, Ab, false, B0, (short)0, ab_acc[0][mt], false, false);
            ab_acc[1][mt] = __builtin_amdgcn_wmma_f32_16x16x4_f32(
                false, Ab, false, B1, (short)0, ab_acc[1][mt], false, false);
        }
    }
    // C/D layout: VGPR r -> row r (lanes 0-15) / row r+8 (lanes 16-31)
#pragma unroll
    for (int b = 0; b < 2; ++b)
#pragma unroll
        for (int mt = 0; mt < 2; ++mt)
#pragma unroll
            for (int r = 0; r < 8; ++r) {
                int m = mt * 16 + half * 8 + r;
                if (m < kC) {
                    ys[((size_t)b * kC + m) * kNPIX + np + lid] = as_acc[b][mt][r];
                    yb[((size_t)b * kC + m) * kNPIX + np + lid] = ab_acc[b][mt][r];
                }
            }
}

// ---------------------------------------------------------------------------
// K4: two 49-tap guided convolutions + q update.
//     The 49x16x16 bilateral-filter tile (50 KB) is gathered into LDS by the
//     Tensor Data Mover: one 3D-tile TENSOR_LOAD_TO_LDS per block
//     (tile = x:16 contig, y:16 stride W, t:49 stride NPIX -> dense in LDS),
//     then reused across all 19 channels.
// ---------------------------------------------------------------------------
__global__ __launch_bounds__(256) void conv_update(
    const float* __restrict__ u, const float* __restrict__ ys, const float* __restrict__ yb,
    const float* __restrict__ fb, const float* __restrict__ inv_ns,
    const float* __restrict__ inv_nb, float* __restrict__ q) {
    __shared__ float s_fb[kTAPS][256];
    __shared__ float s_ys[22 * 22];
    __shared__ float s_yb[22 * 22];
    __shared__ float s_wt[kTAPS];
    __shared__ float s_ins[256];
    __shared__ float s_inb[256];

    const int tid = threadIdx.x;
    const int lx = tid & 15, ly = tid >> 4;
    const int b = blockIdx.z;
    const int th = blockIdx.y * 16, tw = blockIdx.x * 16;
    const int p = (th + ly) * kW + (tw + lx);

    // --- TDM bulk gather of the bilateral-filter tile (wave 0 issues once) ---
    if (tid < 32) {
        unsigned lds_off = (unsigned)(unsigned long long)
            (__attribute__((address_space(3))) float*)&s_fb[0][0];
        unsigned long long ga =
            (unsigned long long)(size_t)(fb + (size_t)b * kTAPS * kNPIX + (size_t)th * kW + tw);
        // D# group0: count=1 | lds_addr | global_addr[56:0] | type=2
        u32x4 g0;
        g0.x = 1u;                                   // count=1, user mode
        g0.y = lds_off;                              // LDS byte address
        g0.z = (unsigned)ga;                         // global_addr[31:0]
        g0.w = (unsigned)((ga >> 32) & 0x01FFFFFFull) | (2u << 30);  // addr[56:32] | type=2
        // D# group1: data_size=4B; tensor_dim0=W, tensor_dim1=H;
        //            tile=(16,16,49); strides=(W, NPIX)
        i32x8 g1;
        g1[0] = 0x20000;                    // workgroup_mask=0, data_size=2 (4B)
        g1[1] = (kW << 16);                 // atomic_barrier_addr=0 | tensor_dim0[15:0]
        g1[2] = (kH << 16);                 // tensor_dim0[31:16]=0 | tensor_dim1[15:0]
        g1[3] = (16 << 16);                 // tensor_dim1[31:16]=0 | tile_dim0=16
        g1[4] = 16 | (kTAPS << 16);         // tile_dim1=16 | tile_dim2=49
        g1[5] = kW;                         // tensor_dim0_stride[31:0] = 160
        g1[6] = (int)((unsigned)kNPIX << 16); // dim0_stride[47:32]=0 | dim1_stride[15:0]
        g1[7] = kNPIX >> 16;                // tensor_dim1_stride[47:16]
        // D# group2: tensor_dim2=49, tensor_dim3=1, dim2_stride=NPIX, tile_dim3=0
        i32x4 g2;
        g2.x = kTAPS; g2.y = 1; g2.z = kNPIX; g2.w = 0;
        i32x4 g3 = {0, 0, 0, 0};
        i32x8 g4 = {0, 0, 0, 0, 0, 0, 0, 0};   // extra descriptor group (clang-23 6-arg form)
        __builtin_amdgcn_tensor_load_to_lds(g0, g1, g2, g3, g4, 0);
        __builtin_amdgcn_s_wait_tensorcnt(0);
    }

    if (tid < kTAPS) {
        int dx = tid / kFS - kSPAN, dy = tid % kFS - kSPAN;
        s_wt[tid] = __expf(-0.5f * (float)(dx * dx + dy * dy) * kINV_BG2);
    }
    s_ins[tid] = inv_ns[p];
    s_inb[tid] = inv_nb[(size_t)b * kNPIX + p];

    for (int c = 0; c < kC; ++c) {
        const size_t base = ((size_t)b * kC + c) * kNPIX;
        for (int i = tid; i < 22 * 22; i += 256) {
            int hh = th - kSPAN + i / 22, ww = tw - kSPAN + i % 22;
            bool ok = (hh >= 0) && (hh < kH) && (ww >= 0) && (ww < kW);
            s_ys[i] = ok ? ys[base + hh * kW + ww] : 0.f;
            s_yb[i] = ok ? yb[base + hh * kW + ww] : 0.f;
        }
        __syncthreads();
        float accs = 0.f, accb = 0.f;
#pragma unroll
        for (int dx7 = 0; dx7 < kFS; ++dx7)
#pragma unroll
            for (int dy7 = 0; dy7 < kFS; ++dy7) {
                int t = dx7 * kFS + dy7;
                float v_s = s_ys[(ly + dx7) * 22 + (lx + dy7)];
                float v_b = s_yb[(ly + dx7) * 22 + (lx + dy7)];
                accs = fmaf(s_wt[t], v_s, accs);
                accb = fmaf(s_fb[t][tid], v_b, accb);
            }
        q[base + p] = u[base + p] - accs * s_ins[tid] - accb * s_inb[tid];
        __syncthreads();
    }
}

// ---------------------------------------------------------------------------
// K5: NCHW q -> NHWC output
// ---------------------------------------------------------------------------
__global__ void write_out(const float* __restrict__ q, float* __restrict__ out) {
    int idx = blockIdx.x * blockDim.x + threadIdx.x;
    if (idx >= kB * kNPIX * kC) return;
    int c = idx % kC, p = (idx / kC) % kNPIX, b = idx / (kC * kNPIX);
    out[idx] = q[((size_t)b * kC + c) * kNPIX + p];
}

extern "C" void kernel_launch(void* const* d_in, const int* in_sizes, int n_in,
                              void* d_out, int out_size, void* d_ws, size_t ws_size,
                              hipStream_t stream) {
    const float* un  = (const float*)d_in[0];
    const float* rgb = (const float*)d_in[1];
    const float* sw  = (const float*)d_in[2];
    const float* bw  = (const float*)d_in[3];
    const float* cm  = (const float*)d_in[4];
    float* ws = (float*)d_ws;
    float* A      = ws + OFF_A;
    float* inv_ns = ws + OFF_INS;
    float* inv_nb = ws + OFF_INB;
    float* fb     = ws + OFF_FB;
    float* u      = ws + OFF_U;
    float* q      = ws + OFF_Q;
    float* ys     = ws + OFF_YS;
    float* yb     = ws + OFF_YB;

    prep_weights<<<1, 256, 0, stream>>>(sw, bw, cm, A);
    init_q<<<(kB * kC * kNPIX + 255) / 256, 256, 0, stream>>>(un, u, q);
    build_filters<<<dim3((kNPIX + 255) / 256, kB), 256, 0, stream>>>(rgb, fb, inv_ns, inv_nb);

    for (int it = 0; it < kNITER; ++it) {
        gemm_softmax<<<kNPIX / 16 / 8, 256, 0, stream>>>(q, A, ys, yb);
        conv_update<<<dim3(kW / 16, kH / 16, kB), 256, 0, stream>>>(u, ys, yb, fb, inv_ns, inv_nb, q);
    }
    write_out<<<(kB * kNPIX * kC + 255) / 256, 256, 0, stream>>>(q, (float*)d_out);
}